// ScaledDotProductAttention_29850022707608
// MI455X (gfx1250) — compile-verified
//
#include <hip/hip_runtime.h>
#include <math.h>

// Problem shape (fixed by reference): B=4, W=64, H=64, C=256, M=32, D=32
//   q: [B,W,H,M,D]  site*1024 floats
//   v: [B,W,H,C,M]  site*8192 floats   (dominant: 512 MiB streamed once)
//   k: [B,W,H,D,1]  site*32   floats
// out: [B,C,W,H]
// Memory-bound: ~594 MiB total -> ~27 us at 23.3 TB/s; compute is ~0.3 GFLOP.

typedef __attribute__((ext_vector_type(2))) float v2f;
typedef __attribute__((ext_vector_type(8))) float v8f;

#define TEMP_INV 0.125f   // 1/TEMPERATURE

__global__ __launch_bounds__(256) void sdpa_softmax_slot_kernel(
    const float* __restrict__ q,
    const float* __restrict__ v,
    const float* __restrict__ k,
    float* __restrict__ out)
{
    const int site = blockIdx.x;            // b*4096 + w*64 + h
    const int b = site >> 12;
    const int w = (site >> 6) & 63;
    const int h = site & 63;

    __shared__ float sAttn[32];

    const int tid  = threadIdx.x;
    const int lane = tid & 31;
    const int wave = tid >> 5;

    // ---- Phase 1 (wave 0 only): attn = softmax_M( (q @ k) / T ), lane == m ----
    if (wave == 0) {
        const float* qrow = q + (size_t)site * 1024 + lane * 32;
        const float* kv   = k + (size_t)site * 32;
        float acc = 0.f;
        #pragma unroll
        for (int d = 0; d < 32; d += 4) {
            float4 qv = *(const float4*)(qrow + d);   // 128b coalesced
            float4 kk = *(const float4*)(kv + d);     // wave-uniform -> broadcast
            acc = fmaf(qv.x, kk.x, acc);
            acc = fmaf(qv.y, kk.y, acc);
            acc = fmaf(qv.z, kk.z, acc);
            acc = fmaf(qv.w, kk.w, acc);
        }
        float logit = acc * TEMP_INV;
        // wave32 softmax across lanes (M == warpSize on gfx1250)
        float mx = logit;
        #pragma unroll
        for (int off = 16; off >= 1; off >>= 1)
            mx = fmaxf(mx, __shfl_xor(mx, off, 32));
        float e = __expf(logit - mx);
        float s = e;
        #pragma unroll
        for (int off = 16; off >= 1; off >>= 1)
            s += __shfl_xor(s, off, 32);
        sAttn[lane] = e / s;
    }
    __syncthreads();

    // ---- Phase 2 (all 8 waves): out[c] = sum_m v[c,m]*attn[m] via
    // V_WMMA_F32_16X16X4_F32.  A[i,k]=attn[4t+k] (rows replicated),
    // B[k,j]=v[cbase+j][4t+k].  D rows are all equal to the matvec result.
    // Each wave covers 2 of the 16 c-tiles.
    const int hi  = (lane >> 4) << 1;       // K-pair owned by this half-wave: 0 or 2
    const int col = lane & 15;              // B column / output c within tile
    const int c0  = (wave * 2) * 16;
    const int c1  = c0 + 16;

    const float* vs = v + (size_t)site * 8192;
    const float* p0 = vs + (size_t)(c0 + col) * 32 + hi;
    const float* p1 = vs + (size_t)(c1 + col) * 32 + hi;

    v8f acc0 = {};
    v8f acc1 = {};
    #pragma unroll
    for (int t = 0; t < 8; ++t) {           // K = 32 = 8 steps of 4
        v2f a;
        a.x = sAttn[4 * t + hi];            // A fragment: lanes0-15 K=0,1; lanes16-31 K=2,3
        a.y = sAttn[4 * t + hi + 1];
        const v2f b0 = *(const v2f*)(p0 + 4 * t);   // global_load_b64, v streamed once
        const v2f b1 = *(const v2f*)(p1 + 4 * t);
        acc0 = __builtin_amdgcn_wmma_f32_16x16x4_f32(
            false, a, false, b0, (short)0, acc0, false, false);
        acc1 = __builtin_amdgcn_wmma_f32_16x16x4_f32(
            false, a, false, b1, (short)0, acc1, false, false);
    }

    // Row 0 of D lives in VGPR0 lanes 0-15; row 8 in lanes 16-31 (identical value).
    // Full-wave store: lanes 0-15 emit tile0, lanes 16-31 emit tile1.
    const float val = (lane < 16) ? acc0[0] : acc1[0];
    const int   c   = ((lane < 16) ? c0 : c1) + col;
    out[(((size_t)b * 256 + c) * 64 + w) * 64 + h] = val;   // L2 absorbs the scatter
}

extern "C" void kernel_launch(void* const* d_in, const int* in_sizes, int n_in,
                              void* d_out, int out_size, void* d_ws, size_t ws_size,
                              hipStream_t stream) {
    const float* q = (const float*)d_in[0];
    const float* v = (const float*)d_in[1];
    const float* k = (const float*)d_in[2];
    float* out = (float*)d_out;
    // one block per (b,w,h) site: 4*64*64 = 16384 blocks, 8 waves each
    sdpa_softmax_slot_kernel<<<dim3(16384), dim3(256), 0, stream>>>(q, v, k, out);
}